// ClusterPooling_59416577573592
// MI455X (gfx1250) — compile-verified
//
#include <hip/hip_runtime.h>
#include <hip/hip_bf16.h>
#include <cstddef>

// ---------------------------------------------------------------------------
// ClusterPooling for MI455X (gfx1250, wave32).
//   n = 8192 nodes, f = 256 feats, E = 131072 edges (2E directed).
// Dense S (n x n) staged in the A_c region of d_out; T = S^T @ x uses
// V_WMMA_F32_16X16X4_F32 with LDS-staged tiles (coalesced b128 global loads,
// global_prefetch on the streamed S chunks). All other reductions are sparse
// scatter/atomic ops. Fixed-iteration CC (graph-capture-safe).
// ---------------------------------------------------------------------------

typedef __attribute__((ext_vector_type(2))) float v2f;
typedef __attribute__((ext_vector_type(8))) float v8f;

#if __has_builtin(__builtin_amdgcn_wmma_f32_16x16x4_f32)
#define HAVE_WMMA_F32 1
#endif

#define CP_FEAT 256
#define CP_KB   32          // K-chunk staged in LDS
#define CP_PADB 8           // pad x rows in LDS: stride 264 -> half-waves hit
                            // disjoint bank groups (264*2 mod 64 == 16)

// ---- init: label[i] = i, deg[i] = 0 ---------------------------------------
__global__ void cp_init(int* __restrict__ label, int* __restrict__ deg, int n) {
  int i = blockIdx.x * blockDim.x + threadIdx.x;
  if (i < n) { label[i] = i; deg[i] = 0; }
}

// ---- grid-stride float fill ----------------------------------------------
__global__ void cp_fill_f32(float* __restrict__ p, size_t count, float v) {
  size_t i = (size_t)blockIdx.x * blockDim.x + threadIdx.x;
  size_t stride = (size_t)gridDim.x * blockDim.x;
  for (; i < count; i += stride) p[i] = v;
}

// ---- p1 = x @ w1, p2 = x @ w2 (one block of 256 per node) -----------------
__global__ __launch_bounds__(CP_FEAT) void cp_gemv2(
    const float* __restrict__ x, const float* __restrict__ w,
    float* __restrict__ p1, float* __restrict__ p2, int f) {
  __shared__ float s1[CP_FEAT];
  __shared__ float s2[CP_FEAT];
  const int i = blockIdx.x;
  const int c = threadIdx.x;
  float xv = x[(size_t)i * f + c];
  s1[c] = xv * w[c];
  s2[c] = xv * w[f + c];
  __syncthreads();
  for (int off = CP_FEAT / 2; off > 0; off >>= 1) {
    if (c < off) { s1[c] += s1[c + off]; s2[c] += s2[c + off]; }
    __syncthreads();
  }
  if (c == 0) { p1[i] = s1[0]; p2[i] = s2[0]; }
}

__device__ __forceinline__ void cp_edge(const int* ei, int E, int t, int& s, int& d) {
  if (t < E) { s = ei[t];           d = ei[E + t]; }
  else       { s = ei[E + (t - E)]; d = ei[t - E]; }
}

// ---- e = tanh(p1[s] + p2[d] + b); deg += contract --------------------------
__global__ void cp_edge_score(const int* __restrict__ ei,
                              const float* __restrict__ p1,
                              const float* __restrict__ p2,
                              const float* __restrict__ lb,
                              float* __restrict__ e,
                              int* __restrict__ deg, int E) {
  int t = blockIdx.x * blockDim.x + threadIdx.x;
  if (t >= 2 * E) return;
  int s, d; cp_edge(ei, E, t, s, d);
  float ev = tanhf(p1[s] + p2[d] + lb[0]);
  e[t] = ev;
  if (s != d && ev > 0.0f) {
    atomicAdd(&deg[s], 1);
    atomicAdd(&deg[d], 1);
  }
}

// ---- CC: min-label relax over contract edges ------------------------------
__global__ void cp_cc_relax(const int* __restrict__ ei,
                            const float* __restrict__ e,
                            int* __restrict__ label, int E) {
  int t = blockIdx.x * blockDim.x + threadIdx.x;
  if (t >= 2 * E) return;
  int s, d; cp_edge(ei, E, t, s, d);
  if (s != d && e[t] > 0.0f) {
    int ls = label[s], ld = label[d];
    int m = ls < ld ? ls : ld;
    if (m < ls) atomicMin(&label[s], m);
    if (m < ld) atomicMin(&label[d], m);
  }
}

// ---- pointer jumping (monotone, safe in-place) ----------------------------
__global__ void cp_cc_jump(int* __restrict__ label, int n) {
  int i = blockIdx.x * blockDim.x + threadIdx.x;
  if (i < n) label[i] = label[label[i]];
}

// ---- is_root cumsum -> cluster (single block, n <= 16384) -----------------
__global__ __launch_bounds__(1024) void cp_scan_cluster(
    const int* __restrict__ label, int* __restrict__ csum,
    int* __restrict__ cluster, int n) {
  __shared__ int part[1024];
  const int t = threadIdx.x;
  const int per = (n + 1023) / 1024;   // 8 for n=8192
  const int base = t * per;
  int loc[16];
  int s = 0;
  for (int i = 0; i < per; ++i) {
    int g = base + i;
    int v = (g < n && label[g] == g) ? 1 : 0;
    s += v;
    loc[i] = s;
  }
  part[t] = s;
  __syncthreads();
  for (int off = 1; off < 1024; off <<= 1) {
    int v = (t >= off) ? part[t - off] : 0;
    __syncthreads();
    part[t] += v;
    __syncthreads();
  }
  int pre = part[t] - s;   // exclusive prefix of this thread's chunk
  for (int i = 0; i < per; ++i) {
    int g = base + i;
    if (g < n) csum[g] = pre + loc[i];
  }
  __syncthreads();
  for (int i = t; i < n; i += 1024) cluster[i] = csum[label[i]] - 1;
}

// ---- S[s,d] += e over masked edges (S lives in A_c slot of d_out) ---------
__global__ void cp_scatter_S(const int* __restrict__ ei,
                             const float* __restrict__ e,
                             float* __restrict__ S, int E, int n) {
  int t = blockIdx.x * blockDim.x + threadIdx.x;
  if (t >= 2 * E) return;
  int s, d; cp_edge(ei, E, t, s, d);
  if (s != d) atomicAdd(&S[(size_t)s * n + d], e[t]);
}

// ---- S[i,i] = 1 for isolated nodes ----------------------------------------
__global__ void cp_s_diag(float* __restrict__ S, const int* __restrict__ deg, int n) {
  int i = blockIdx.x * blockDim.x + threadIdx.x;
  if (i < n && deg[i] == 0) S[(size_t)i * n + i] = 1.0f;
}

// ---- T = S^T @ x via V_WMMA_F32_16X16X4_F32, LDS-staged tiles -------------
// Grid: n/16 blocks (j-tiles), 512 threads = 16 waves; wave w -> c-tile w*16.
// Per K-chunk of 32: stage S[k:k+32, j0:j0+16] (2KB, shared by all 16 waves)
// and x[k:k+32, 0:256] (33KB, b128 coalesced rows, +8-float row pad for LDS
// banking). fp32 WMMA fragment layout: lanes 0-15 hold K pair {0,1},
// lanes 16-31 hold {2,3}; M/N = lane%16.
__global__ __launch_bounds__(512) void cp_gemm_stx(
    const float* __restrict__ S, const float* __restrict__ x,
    float* __restrict__ T, int n, int f) {
  __shared__ float lA[CP_KB * 16];
  __shared__ float lB[CP_KB * (CP_FEAT + CP_PADB)];
  const int tid  = threadIdx.x;
  const int wave = tid >> 5;
  const int lane = tid & 31;
  const int h = lane >> 4;        // half: 0 or 1
  const int l = lane & 15;
  const int j0 = blockIdx.x * 16; // output row tile (j dimension)
  const int c0 = wave * 16;       // output col tile (feature dimension)
  const int ra = tid >> 4;        // 0..31 : row within chunk (A and B loads)
  const int ca = tid & 15;
#if defined(HAVE_WMMA_F32)
  v8f acc = {0.f, 0.f, 0.f, 0.f, 0.f, 0.f, 0.f, 0.f};
  for (int k = 0; k < n; k += CP_KB) {
    // Stage A chunk: S[k+ra, j0+ca] -- 64B contiguous per 16-thread group.
    lA[ra * 16 + ca] = S[(size_t)(k + ra) * n + j0 + ca];
    // Prefetch next S chunk while this one computes (streams 256MB once).
    if (k + CP_KB < n)
      __builtin_prefetch(&S[(size_t)(k + CP_KB + ra) * n + j0 + ca], 0, 1);
    // Stage B chunk: x[k+ra, ca*16 .. ca*16+15] -- 4x b128, 1KB rows.
    {
      const float4* src = (const float4*)(x + (size_t)(k + ra) * f + ca * 16);
      float4* dst = (float4*)(lB + ra * (CP_FEAT + CP_PADB) + ca * 16);
      dst[0] = src[0]; dst[1] = src[1]; dst[2] = src[2]; dst[3] = src[3];
    }
    __syncthreads();
#pragma unroll
    for (int kk = 0; kk < CP_KB; kk += 4) {
      const int ka = kk + h * 2;
      v2f a, b;
      a.x = lA[(ka + 0) * 16 + l];
      a.y = lA[(ka + 1) * 16 + l];
      b.x = lB[(ka + 0) * (CP_FEAT + CP_PADB) + c0 + l];
      b.y = lB[(ka + 1) * (CP_FEAT + CP_PADB) + c0 + l];
      acc = __builtin_amdgcn_wmma_f32_16x16x4_f32(
          /*neg_a=*/false, a, /*neg_b=*/false, b,
          /*c_mod=*/(short)0, acc, /*reuse_a=*/false, /*reuse_b=*/false);
    }
    __syncthreads();
  }
#pragma unroll
  for (int r = 0; r < 8; ++r) {
    // C/D layout: VGPR r -> row M = r + 8*h, col N = l
    T[(size_t)(j0 + r + h * 8) * f + c0 + l] = acc[r];
  }
#else
  // Scalar fallback (keeps compile alive if builtin probe regresses).
  float facc[8] = {0.f};
  for (int k = 0; k < n; ++k) {
    float bx = x[(size_t)k * f + c0 + l];
    #pragma unroll
    for (int r = 0; r < 8; ++r)
      facc[r] += S[(size_t)k * n + j0 + r + h * 8] * bx;
  }
  #pragma unroll
  for (int r = 0; r < 8; ++r)
    T[(size_t)(j0 + r + h * 8) * f + c0 + l] = facc[r];
#endif
}

// ---- X_new[cluster[j], :] += T[j, :] --------------------------------------
__global__ void cp_seg_x(const float* __restrict__ T,
                         const int* __restrict__ cluster,
                         float* __restrict__ Xn, int n, int f) {
  size_t idx = (size_t)blockIdx.x * blockDim.x + threadIdx.x;
  if (idx >= (size_t)n * f) return;
  int j = (int)(idx / f);
  int c = (int)(idx % f);
  atomicAdd(&Xn[(size_t)cluster[j] * f + c], T[idx]);
}

// ---- A_c[cluster[s], cluster[d]] += 1 over masked edges -------------------
__global__ void cp_scatter_Ac(const int* __restrict__ ei,
                              const int* __restrict__ cluster,
                              float* __restrict__ Ac, int E, int n) {
  int t = blockIdx.x * blockDim.x + threadIdx.x;
  if (t >= 2 * E) return;
  int s, d; cp_edge(ei, E, t, s, d);
  if (s != d) atomicAdd(&Ac[(size_t)cluster[s] * n + cluster[d]], 1.0f);
}

__global__ void cp_zero_diag(float* __restrict__ Ac, int n) {
  int i = blockIdx.x * blockDim.x + threadIdx.x;
  if (i < n) Ac[(size_t)i * n + i] = 0.0f;
}

// ---- new_batch[cluster[i]] = batch[i]; cluster output as float ------------
__global__ void cp_finalize(const int* __restrict__ cluster,
                            const int* __restrict__ batch,
                            float* __restrict__ nb,
                            float* __restrict__ clo, int n) {
  int i = blockIdx.x * blockDim.x + threadIdx.x;
  if (i < n) {
    nb[cluster[i]] = (float)batch[i];
    clo[i] = (float)cluster[i];
  }
}

extern "C" void kernel_launch(void* const* d_in, const int* in_sizes, int n_in,
                              void* d_out, int out_size, void* d_ws, size_t ws_size,
                              hipStream_t stream) {
  const float* x     = (const float*)d_in[0];
  const int*   ei    = (const int*)d_in[1];
  const int*   batch = (const int*)d_in[2];
  const float* lw    = (const float*)d_in[3];
  const float* lb    = (const float*)d_in[4];

  const int f  = CP_FEAT;
  const int n  = in_sizes[0] / f;     // 8192
  const int E  = in_sizes[1] / 2;     // 131072
  const int E2 = 2 * E;

  // ---- workspace layout (~10 MB) ----
  float* p1   = (float*)d_ws;                    // n
  float* p2   = p1 + n;                          // n
  float* ebuf = p2 + n;                          // 2E
  float* T    = ebuf + E2;                       // n*f
  int* label  = (int*)(T + (size_t)n * f);       // n
  int* deg    = label + n;                       // n
  int* csum   = deg + n;                         // n
  int* cluster = csum + n;                       // n

  // ---- output layout ----
  float* Xn  = (float*)d_out;                    // n*f
  float* Ac  = Xn + (size_t)n * f;               // n*n (also S staging)
  float* nb  = Ac + (size_t)n * n;               // n
  float* clo = nb + n;                           // n

  const int B = 256;
  const int gN  = (n + B - 1) / B;
  const int gE2 = (E2 + B - 1) / B;

  // 1) init + edge scores + degrees
  cp_init<<<gN, B, 0, stream>>>(label, deg, n);
  cp_gemv2<<<n, CP_FEAT, 0, stream>>>(x, lw, p1, p2, f);
  cp_edge_score<<<gE2, B, 0, stream>>>(ei, p1, p2, lb, ebuf, deg, E);

  // 2) connected components: fixed 40 rounds (capture-safe, converged)
  for (int it = 0; it < 40; ++it) {
    cp_cc_relax<<<gE2, B, 0, stream>>>(ei, ebuf, label, E);
    cp_cc_jump<<<gN, B, 0, stream>>>(label, n);
    cp_cc_jump<<<gN, B, 0, stream>>>(label, n);
  }

  // 3) compact labels -> cluster ids
  cp_scan_cluster<<<1, 1024, 0, stream>>>(label, csum, cluster, n);

  // 4) build dense S in the A_c slot, then T = S^T @ x with fp32 WMMA
  cp_fill_f32<<<8192, B, 0, stream>>>(Ac, (size_t)n * n, 0.0f);
  cp_scatter_S<<<gE2, B, 0, stream>>>(ei, ebuf, Ac, E, n);
  cp_s_diag<<<gN, B, 0, stream>>>(Ac, deg, n);
  cp_gemm_stx<<<n / 16, 512, 0, stream>>>(Ac, x, T, n, f);

  // 5) X_new = segment_sum(T, cluster)
  cp_fill_f32<<<2048, B, 0, stream>>>(Xn, (size_t)n * f, 0.0f);
  cp_seg_x<<<(int)(((size_t)n * f + B - 1) / B), B, 0, stream>>>(T, cluster, Xn, n, f);

  // 6) A_c = edge-count scatter in cluster space, zero diagonal
  cp_fill_f32<<<8192, B, 0, stream>>>(Ac, (size_t)n * n, 0.0f);
  cp_scatter_Ac<<<gE2, B, 0, stream>>>(ei, cluster, Ac, E, n);
  cp_zero_diag<<<gN, B, 0, stream>>>(Ac, n);

  // 7) new_batch + cluster outputs
  cp_fill_f32<<<gN, B, 0, stream>>>(nb, (size_t)n, 0.0f);
  cp_finalize<<<gN, B, 0, stream>>>(cluster, batch, nb, clo, n);
}